// CrossModalCoherenceMetric_61615600828749
// MI455X (gfx1250) — compile-verified
//
#include <hip/hip_runtime.h>
#include <hip/hip_bf16.h>

// ---------------------------------------------------------------------------
// CrossModalCoherenceMetric on MI455X (gfx1250, wave32, WMMA)
//
// N = 8192 steps, M = 4096 modal / 4096 negative, D = 1024, fp32.
// Core math: fp32 GEMMs via V_WMMA_F32_16X16X4_F32 (native f32 matrix path).
//
// Blocking: each block owns a 32-row step strip (two 16-row WMMA strips,
// 128 KB LDS of CDNA5's 320 KB/WGP). Per k-step a wave does one global B
// fragment load shared by TWO WMMAs into alternating accumulators:
//   - halves B-operand L2 traffic (the roofline limiter),
//   - breaks the WMMA->WMMA same-accumulator RAW chain.
//
// Two GEMM passes total: step x modal (stats + raw-G spill to d_out) and
// step x negatives (row-max only). Attention is finalized by a streaming
// elementwise pass over the spilled G (268 MB HBM ~ 12 us @ 23.3 TB/s),
// replacing an entire 68.7 GFLOP recompute GEMM.
// ---------------------------------------------------------------------------

typedef __attribute__((ext_vector_type(2))) float v2f;
typedef __attribute__((ext_vector_type(8))) float v8f;

#define NROWS   8192
#define NCOLS   4096
#define KDIM    1024
#define INV_T   (1.0f / 0.07f)

// d_out flat layout (fp32), total 33562632 elements.
#define ATT_OFF       2ULL
#define ATT_N         (8192ULL * 4096ULL)
#define SCALAR_OFF    (ATT_OFF + ATT_N)          // 33554434
#define PER_STEP_OFF  (SCALAR_OFF + 4ULL)        // 33554438
#define MIN_OFF       (PER_STEP_OFF + 8192ULL)   // 33562630
#define OVR_OFF       (MIN_OFF + 1ULL)           // 33562631

// d_ws float layout.
#define WS_NS   0
#define WS_NM   8192
#define WS_NN   12288
#define WS_LM   16384
#define WS_Z    24576
#define WS_W    32768
#define WS_MSN  40960

// ---------------------------------------------------------------------------
// Kernel 1: clipped row L2 norms for all three matrices. One wave per row.
// ---------------------------------------------------------------------------
__global__ __launch_bounds__(256) void norms_kernel(const float* __restrict__ step,
                                                    const float* __restrict__ modal,
                                                    const float* __restrict__ neg,
                                                    float* __restrict__ ws) {
    const int lane = threadIdx.x & 31;
    const int row  = blockIdx.x * 8 + (threadIdx.x >> 5);  // 16384 rows total
    const float* src;
    if (row < 8192)        src = step  + (size_t)row * KDIM;
    else if (row < 12288)  src = modal + (size_t)(row - 8192) * KDIM;
    else                   src = neg   + (size_t)(row - 12288) * KDIM;

    float s = 0.0f;
#pragma unroll
    for (int i = 0; i < 8; ++i) {
        float4 x = *(const float4*)(src + i * 128 + lane * 4);
        s += x.x * x.x + x.y * x.y + x.z * x.z + x.w * x.w;
    }
#pragma unroll
    for (int m = 16; m >= 1; m >>= 1) s += __shfl_xor(s, m, 32);
    if (lane == 0) ws[row] = fmaxf(sqrtf(s), 1e-8f);
}

// ---------------------------------------------------------------------------
// Cooperative, bank-conflict-free swizzled load of a 32-row A strip into LDS.
// Chunk c (4 floats) of row r is stored at chunk (c ^ (r & 15)): the 16 lanes
// reading 16 rows of one strip at the same k hit 16 distinct 4-bank groups.
// ---------------------------------------------------------------------------
__device__ __forceinline__ void load_a_strip32(float* __restrict__ lds,
                                               const float* __restrict__ A,
                                               int i0, int tid) {
    const int r  = tid & 31;
    const int cg = tid >> 5;  // 0..3
    const float4* src = (const float4*)(A + (size_t)(i0 + r) * KDIM);
#pragma unroll 4
    for (int i = 0; i < 64; ++i) {
        const int c = cg * 64 + i;
        *(float4*)(&lds[r * KDIM + ((c ^ (r & 15)) << 2)]) = src[c];
    }
}

// Two 16x16 fp32 tiles (rows i0..i0+15 and i0+16..i0+31) sharing one B
// fragment stream; alternating accumulators keep the XDL pipe free of
// back-to-back same-accumulator RAW hazards.
__device__ __forceinline__ void gemm_tile2(const float* __restrict__ lds,
                                           const float* __restrict__ bptr,
                                           int ln, int half,
                                           v8f& acc0, v8f& acc1) {
    acc0 = (v8f){};
    acc1 = (v8f){};
    for (int k = 0; k < KDIM; k += 4) {
        const int coff = (((k >> 2) ^ ln) << 2) + half * 2;
        // A frags: lane holds rows (lane&15) and (lane&15)+16, K = k + (lane>>4)*2 + {0,1}
        v2f a0 = *(const v2f*)(&lds[(ln << 10) + coff]);
        v2f a1 = *(const v2f*)(&lds[((ln + 16) << 10) + coff]);
        // B frag: lane holds col (lane&15), K = k + (lane>>4)*2 + {0,1}
        v2f b = *(const v2f*)(bptr + k);
        acc0 = __builtin_amdgcn_wmma_f32_16x16x4_f32(false, a0, false, b,
                                                     (short)0, acc0, false, false);
        acc1 = __builtin_amdgcn_wmma_f32_16x16x4_f32(false, a1, false, b,
                                                     (short)0, acc1, false, false);
    }
}

// ---------------------------------------------------------------------------
// Kernel 2: fused GEMM + row statistics over a 32-row strip.
//   FULL=true  (modal):     spills raw G to d_out's attention region, and
//                           tracks per-row max cosine (-> per_step), online
//                           softmax max/Z, and sum(S * e).
//   FULL=false (negatives): per-row max cosine only.
// 256 blocks x 128 threads; each of the 4 waves sweeps 64 of the 256 column
// tiles. EXEC is all-ones through every WMMA region.
// ---------------------------------------------------------------------------
template <bool FULL>
__global__ __launch_bounds__(128) void stats_kernel(const float* __restrict__ A,
                                                    const float* __restrict__ B,
                                                    const float* __restrict__ normA,
                                                    const float* __restrict__ normB,
                                                    float* __restrict__ gout,
                                                    float* __restrict__ rowmax_out,
                                                    float* __restrict__ lmax_out,
                                                    float* __restrict__ z_out,
                                                    float* __restrict__ w_out) {
    __shared__ __align__(16) float lds[32 * KDIM];  // 128 KB A strip (reused for partials)
    const int tid  = threadIdx.x;
    const int lane = tid & 31;
    const int wave = tid >> 5;
    const int half = lane >> 4;
    const int ln   = lane & 15;
    const int i0   = blockIdx.x * 32;

    load_a_strip32(lds, A, i0, tid);
    __syncthreads();

    // Per-lane reciprocal step norms; index sv = strip*8 + v maps to
    // global row i0 + strip*16 + half*8 + v.
    float rns[16];
#pragma unroll
    for (int s = 0; s < 2; ++s)
#pragma unroll
        for (int v = 0; v < 8; ++v)
            rns[s * 8 + v] = 1.0f / normA[i0 + s * 16 + half * 8 + v];

    float mS[16], mL[16], Zl[16], Wl[16];
#pragma unroll
    for (int i = 0; i < 16; ++i) { mS[i] = -3.0e38f; mL[i] = -3.0e38f; Zl[i] = 0.0f; Wl[i] = 0.0f; }

    for (int t = 0; t < 64; ++t) {
        const int j0 = (wave + 4 * t) << 4;
        const float rnb = 1.0f / normB[j0 + ln];
        const float* bptr = B + (size_t)(j0 + ln) * KDIM + half * 2;
        if (t < 63)  // warm caches for next tile's B rows (global_prefetch_b8)
            __builtin_prefetch(B + (size_t)(j0 + 64 + ln) * KDIM, 0, 0);

        v8f acc0, acc1;
        gemm_tile2(lds, bptr, ln, half, acc0, acc1);

#pragma unroll
        for (int s = 0; s < 2; ++s) {
#pragma unroll
            for (int v = 0; v < 8; ++v) {
                const int sv = s * 8 + v;
                const float g = s == 0 ? acc0[v] : acc1[v];
                if (FULL)  // spill raw logit input for the attention finalize pass
                    gout[(size_t)(i0 + s * 16 + half * 8 + v) * NCOLS + j0 + ln] = g;
                const float S = g * rns[sv] * rnb;
                mS[sv] = fmaxf(mS[sv], S);
                if (FULL) {
                    const float l    = g * INV_T;
                    const float mNew = fmaxf(mL[sv], l);
                    const float sc   = __expf(mL[sv] - mNew);
                    const float e    = __expf(l - mNew);
                    Zl[sv] = Zl[sv] * sc + e;
                    Wl[sv] = Wl[sv] * sc + S * e;
                    mL[sv] = mNew;
                }
            }
        }
    }

    // Butterfly reduce across the 16 lanes that share each row.
#pragma unroll
    for (int m = 1; m <= 8; m <<= 1) {
#pragma unroll
        for (int sv = 0; sv < 16; ++sv) {
            mS[sv] = fmaxf(mS[sv], __shfl_xor(mS[sv], m, 32));
            if (FULL) {
                const float m2 = __shfl_xor(mL[sv], m, 32);
                const float z2 = __shfl_xor(Zl[sv], m, 32);
                const float w2 = __shfl_xor(Wl[sv], m, 32);
                const float mNew = fmaxf(mL[sv], m2);
                const float s1 = __expf(mL[sv] - mNew), s2 = __expf(m2 - mNew);
                Zl[sv] = Zl[sv] * s1 + z2 * s2;
                Wl[sv] = Wl[sv] * s1 + w2 * s2;
                mL[sv] = mNew;
            }
        }
    }

    __syncthreads();  // A strip dead; reuse LDS for per-wave partials
    float* pS = lds;
    float* pM = lds + 128;
    float* pZ = lds + 256;
    float* pW = lds + 384;
    if (ln == 0) {  // lanes 0 and 16 each publish 16 rows
#pragma unroll
        for (int s = 0; s < 2; ++s)
#pragma unroll
            for (int v = 0; v < 8; ++v) {
                const int sv = s * 8 + v;
                const int r  = wave * 32 + s * 16 + half * 8 + v;
                pS[r] = mS[sv];
                if (FULL) { pM[r] = mL[sv]; pZ[r] = Zl[sv]; pW[r] = Wl[sv]; }
            }
    }
    __syncthreads();
    if (tid < 32) {
        float m = -3.0e38f, ml = -3.0e38f, z = 0.0f, w = 0.0f;
#pragma unroll
        for (int wv = 0; wv < 4; ++wv) {
            const int r = wv * 32 + tid;
            m = fmaxf(m, pS[r]);
            if (FULL) {
                const float m2 = pM[r], z2 = pZ[r], w2 = pW[r];
                const float mNew = fmaxf(ml, m2);
                const float s1 = __expf(ml - mNew), s2 = __expf(m2 - mNew);
                z = z * s1 + z2 * s2;
                w = w * s1 + w2 * s2;
                ml = mNew;
            }
        }
        rowmax_out[i0 + tid] = m;
        if (FULL) { lmax_out[i0 + tid] = ml; z_out[i0 + tid] = z; w_out[i0 + tid] = w; }
    }
}

// ---------------------------------------------------------------------------
// Kernel 3: streaming attention finalize. Reads the spilled raw G (in the
// attention region of d_out) and rewrites it in place as
//   exp(g/T - rowmax) * (1/Z).
// Pure HBM-bandwidth pass: 268 MB r+w ~ 12 us at 23.3 TB/s. One block per
// row; row scalars are block-uniform (scalar loads). float2 (b64) accesses
// because the region starts at d_out+2 (8-byte aligned).
// ---------------------------------------------------------------------------
__global__ __launch_bounds__(256) void attention_finalize_kernel(
        float* __restrict__ att,
        const float* __restrict__ lmax,
        const float* __restrict__ z) {
    const int row = blockIdx.x;
    const float lm = lmax[row];
    const float iz = 1.0f / z[row];
    float2* p = (float2*)(att + (size_t)row * NCOLS);
#pragma unroll
    for (int it = 0; it < (NCOLS / 2) / 256; ++it) {
        const int c = it * 256 + threadIdx.x;
        float2 g = p[c];
        g.x = __expf(g.x * INV_T - lm) * iz;
        g.y = __expf(g.y * INV_T - lm) * iz;
        p[c] = g;
    }
}

// ---------------------------------------------------------------------------
// Kernel 4: final scalar reductions + tuple scalars.
// ---------------------------------------------------------------------------
__global__ __launch_bounds__(256) void finalize_kernel(const float* __restrict__ ws,
                                                       float* __restrict__ out) {
    const float* Z   = ws + WS_Z;
    const float* W   = ws + WS_W;
    const float* mSn = ws + WS_MSN;
    const float* per_step = out + PER_STEP_OFF;

    const int tid = threadIdx.x;
    float sumA = 0.0f, sumW = 0.0f, sumN = 0.0f, mn = 3.0e38f;
    for (int i = tid; i < NROWS; i += 256) {
        const float ps = per_step[i];
        sumA += ps;
        mn = fminf(mn, ps);
        sumW += W[i] / Z[i];
        sumN += mSn[i];
    }
#pragma unroll
    for (int m = 16; m >= 1; m >>= 1) {
        sumA += __shfl_xor(sumA, m, 32);
        sumW += __shfl_xor(sumW, m, 32);
        sumN += __shfl_xor(sumN, m, 32);
        mn = fminf(mn, __shfl_xor(mn, m, 32));
    }
    __shared__ float red[4][8];
    const int lane = tid & 31, wv = tid >> 5;
    if (lane == 0) { red[0][wv] = sumA; red[1][wv] = sumW; red[2][wv] = sumN; red[3][wv] = mn; }
    __syncthreads();
    if (tid == 0) {
        float a = 0.0f, w = 0.0f, n = 0.0f, m = 3.0e38f;
#pragma unroll
        for (int i = 0; i < 8; ++i) {
            a += red[0][i]; w += red[1][i]; n += red[2][i]; m = fminf(m, red[3][i]);
        }
        const float alignment = a * (1.0f / NROWS);
        const float weighted  = w * (1.0f / NROWS);
        const float negA      = n * (1.0f / NROWS);
        const float contrast  = alignment - negA;
        out[0] = alignment;
        out[1] = weighted;
        out[SCALAR_OFF + 0] = contrast;
        out[SCALAR_OFF + 1] = fmaxf(contrast - 0.2f, 0.0f);  // margin_score
        out[SCALAR_OFF + 2] = alignment;                     // positive_alignment
        out[SCALAR_OFF + 3] = negA;                          // negative_alignment
        out[MIN_OFF] = m;                                    // min_step_coherence
        out[OVR_OFF] = 0.7f * weighted + 0.3f * contrast;    // overall
    }
}

// ---------------------------------------------------------------------------
extern "C" void kernel_launch(void* const* d_in, const int* in_sizes, int n_in,
                              void* d_out, int out_size, void* d_ws, size_t ws_size,
                              hipStream_t stream) {
    const float* step  = (const float*)d_in[0];
    const float* modal = (const float*)d_in[1];
    const float* neg   = (const float*)d_in[2];
    float* out = (float*)d_out;
    float* ws  = (float*)d_ws;

    float* ns   = ws + WS_NS;
    float* nm   = ws + WS_NM;
    float* nn   = ws + WS_NN;
    float* lmax = ws + WS_LM;
    float* z    = ws + WS_Z;
    float* w    = ws + WS_W;
    float* msn  = ws + WS_MSN;
    float* per_step = out + PER_STEP_OFF;
    float* att      = out + ATT_OFF;

    // 1) row norms for all 16384 rows
    norms_kernel<<<2048, 256, 0, stream>>>(step, modal, neg, ws);
    // 2) step x modal: raw-G spill + per_step row-max cosine + softmax stats
    stats_kernel<true><<<NROWS / 32, 128, 0, stream>>>(step, modal, ns, nm, att,
                                                       per_step, lmax, z, w);
    // 3) step x negatives: row-max cosine only
    stats_kernel<false><<<NROWS / 32, 128, 0, stream>>>(step, neg, ns, nn, nullptr,
                                                        msn, nullptr, nullptr, nullptr);
    // 4) attention finalize: in-place streaming exp-normalize of spilled G
    attention_finalize_kernel<<<NROWS, 256, 0, stream>>>(att, lmax, z);
    // 5) scalars
    finalize_kernel<<<1, 256, 0, stream>>>(ws, out);
}